// BaseModel_6305011991031
// MI455X (gfx1250) — compile-verified
//
#include <hip/hip_runtime.h>

// Fixed by the reference problem: DIM = 192 (divisible by 4 -> float4 groups
// never straddle a token row: 192/4 = 48 groups per token).
#define DIM   192
#define DIM4  (DIM / 4)

typedef __attribute__((ext_vector_type(4))) float v4f;

// ---------------------------------------------------------------------------
// K0: zero a u32 span (used for tokcnt, entcnt, and d_out accumulator).
// ---------------------------------------------------------------------------
__global__ void segmean_zero_u32(unsigned* __restrict__ p, unsigned n) {
    unsigned stride = gridDim.x * blockDim.x;
    for (unsigned i = blockIdx.x * blockDim.x + threadIdx.x; i < n; i += stride)
        p[i] = 0u;
}

// ---------------------------------------------------------------------------
// K1: tokens-per-mention histogram. 1M u32 atomics into a 0.8MB table (L2).
// ---------------------------------------------------------------------------
__global__ void segmean_count_tokens(const int* __restrict__ t2m,
                                     unsigned* __restrict__ tokcnt,
                                     unsigned n_tokens) {
    unsigned stride = gridDim.x * blockDim.x;
    for (unsigned i = blockIdx.x * blockDim.x + threadIdx.x; i < n_tokens; i += stride)
        atomicAdd(&tokcnt[(unsigned)t2m[i]], 1u);
}

// ---------------------------------------------------------------------------
// K2: per-mention pass: counts -> reciprocals (so the big pass is mul-only),
//     and mentions-per-entity histogram (every mention counts, even empty,
//     matching the reference's segment_sum(ones) semantics).
// ---------------------------------------------------------------------------
__global__ void segmean_mentions(const int* __restrict__ m2e,
                                 const unsigned* __restrict__ tokcnt,
                                 float* __restrict__ winv,
                                 unsigned* __restrict__ entcnt,
                                 unsigned n_mentions) {
    unsigned stride = gridDim.x * blockDim.x;
    for (unsigned i = blockIdx.x * blockDim.x + threadIdx.x; i < n_mentions; i += stride) {
        atomicAdd(&entcnt[(unsigned)m2e[i]], 1u);
        unsigned c = tokcnt[i];
        winv[i] = 1.0f / (float)(c > 0u ? c : 1u);
    }
}

// ---------------------------------------------------------------------------
// K3: the bandwidth-dominant pass. One float4 (b128, non-temporal: streamed
//     exactly once, don't pollute L2 where the 38MB accumulator lives) per
//     thread-iteration; scale by per-mention reciprocal; 4 relaxed agent-scope
//     f32 atomics straight into d_out (entity sums). All 48 groups of a token
//     read the same t2m/winv/m2e words -> broadcast hits.
// ---------------------------------------------------------------------------
__global__ void segmean_scatter(const float* __restrict__ enc,
                                const int* __restrict__ t2m,
                                const int* __restrict__ m2e,
                                const float* __restrict__ winv,
                                float* __restrict__ out,
                                unsigned n_groups /* n_tokens * DIM4 */) {
    unsigned stride = gridDim.x * blockDim.x;
    for (unsigned g = blockIdx.x * blockDim.x + threadIdx.x; g < n_groups; g += stride) {
        unsigned t  = g / DIM4;            // token id
        unsigned d0 = (g - t * DIM4) * 4;  // first dim of this float4
        unsigned m  = (unsigned)t2m[t];
        float    w  = winv[m];
        unsigned e  = (unsigned)m2e[m];
        v4f v = __builtin_nontemporal_load((const v4f*)(enc + (size_t)g * 4));
        float* dst = out + (size_t)e * DIM + d0;
        (void)__hip_atomic_fetch_add(dst + 0, v.x * w, __ATOMIC_RELAXED, __HIP_MEMORY_SCOPE_AGENT);
        (void)__hip_atomic_fetch_add(dst + 1, v.y * w, __ATOMIC_RELAXED, __HIP_MEMORY_SCOPE_AGENT);
        (void)__hip_atomic_fetch_add(dst + 2, v.z * w, __ATOMIC_RELAXED, __HIP_MEMORY_SCOPE_AGENT);
        (void)__hip_atomic_fetch_add(dst + 3, v.w * w, __ATOMIC_RELAXED, __HIP_MEMORY_SCOPE_AGENT);
    }
}

// ---------------------------------------------------------------------------
// K4: finalize: divide each entity row by its mention count (in place).
// ---------------------------------------------------------------------------
__global__ void segmean_finalize(float* __restrict__ out,
                                 const unsigned* __restrict__ entcnt,
                                 unsigned n_groups /* n_entities * DIM4 */) {
    unsigned stride = gridDim.x * blockDim.x;
    for (unsigned g = blockIdx.x * blockDim.x + threadIdx.x; g < n_groups; g += stride) {
        unsigned e = g / DIM4;
        unsigned c = entcnt[e];
        float s = 1.0f / (float)(c > 0u ? c : 1u);
        v4f v = *(v4f*)(out + (size_t)g * 4);
        v *= s;
        *(v4f*)(out + (size_t)g * 4) = v;
    }
}

// ---------------------------------------------------------------------------
// Host-side launch. Inputs: enc_seq [N*D f32], token2mention [N i32],
// mention2entity [M i32], num_mentions (scalar), num_entities (scalar).
// Output: entities [E*D f32]. Workspace: tokcnt[M u32] | winv[M f32] |
// entcnt[E u32]  => ~1.8 MB.
// ---------------------------------------------------------------------------
extern "C" void kernel_launch(void* const* d_in, const int* in_sizes, int n_in,
                              void* d_out, int out_size, void* d_ws, size_t ws_size,
                              hipStream_t stream) {
    const float* enc = (const float*)d_in[0];
    const int*   t2m = (const int*)d_in[1];
    const int*   m2e = (const int*)d_in[2];

    const unsigned n_tokens   = (unsigned)in_sizes[1];
    const unsigned n_mentions = (unsigned)in_sizes[2];
    const unsigned n_entities = (unsigned)(out_size / DIM);

    float* out = (float*)d_out;

    unsigned* tokcnt = (unsigned*)d_ws;
    float*    winv   = (float*)((char*)d_ws + (size_t)n_mentions * 4);
    unsigned* entcnt = (unsigned*)((char*)d_ws + (size_t)n_mentions * 8);

    const int BLK = 256;
    auto blocks = [&](unsigned n) { return (int)((n + BLK - 1) / BLK); };

    // K0: zero accumulators (d_out is an atomic accumulator; ws counters too).
    segmean_zero_u32<<<blocks(n_mentions), BLK, 0, stream>>>(tokcnt, n_mentions);
    segmean_zero_u32<<<blocks(n_entities), BLK, 0, stream>>>(entcnt, n_entities);
    segmean_zero_u32<<<blocks((unsigned)out_size), BLK, 0, stream>>>((unsigned*)out,
                                                                     (unsigned)out_size);

    // K1: token histogram.
    segmean_count_tokens<<<blocks(n_tokens), BLK, 0, stream>>>(t2m, tokcnt, n_tokens);

    // K2: mention reciprocals + entity histogram.
    segmean_mentions<<<blocks(n_mentions), BLK, 0, stream>>>(m2e, tokcnt, winv, entcnt,
                                                             n_mentions);

    // K3: main streaming scatter (768 MB NT read, L2-resident atomics).
    unsigned n_groups = n_tokens * DIM4;
    segmean_scatter<<<blocks(n_groups), BLK, 0, stream>>>(enc, t2m, m2e, winv, out, n_groups);

    // K4: finalize entity means.
    unsigned e_groups = n_entities * DIM4;
    segmean_finalize<<<blocks(e_groups), BLK, 0, stream>>>(out, entcnt, e_groups);
}